// M3GNet_39891656245698
// MI455X (gfx1250) — compile-verified
//
#include <hip/hip_runtime.h>
#include <hip/hip_bf16.h>
#include <cstddef>

#define A_N   50000
#define E_N   1200000
#define T_N   2000000
#define G_N   500
#define U_N   64
#define RBF_N 48

typedef __attribute__((ext_vector_type(2))) float v2f;
typedef __attribute__((ext_vector_type(8))) float v8f;

__device__ __forceinline__ v8f wmma4(v2f a, v2f b, v8f c) {
  // V_WMMA_F32_16X16X4_F32 : D = A(16x4,f32) x B(4x16,f32) + C(16x16,f32)
  return __builtin_amdgcn_wmma_f32_16x16x4_f32(false, a, false, b, (short)0, c,
                                               false, false);
}
__device__ __forceinline__ float sig_(float x)  { return 1.0f / (1.0f + __expf(-x)); }
__device__ __forceinline__ float silu_(float x) { return x * sig_(x); }

// Stage W (K x N row-major, global) into LDS transposed+padded: wt[nc*KP + k].
// KP = K+4 so that bank = (4*nc + k0) % 64 -> 32 lanes of a b64 read cover all
// 64 banks conflict-free.
__device__ __forceinline__ void stage_wt(const float* __restrict__ W, float* __restrict__ wt,
                                         int K, int N, int KP) {
  for (int idx = threadIdx.x; idx < K * N; idx += blockDim.x) {
    const int k = idx / N, nc = idx - k * N;
    wt[nc * KP + k] = W[idx];
  }
}

// ---------------------------------------------------------------- embedding
__global__ void embed_kernel(const int* __restrict__ at, const float* __restrict__ emb,
                             float* __restrict__ atom) {
  int idx = blockIdx.x * blockDim.x + threadIdx.x;   // A*64 threads, exact
  int a = idx >> 6, u = idx & 63;
  atom[idx] = emb[(size_t)at[a] * U_N + u];
}

// ------------------------------------------------------- bond = silu(rbf@W+b)
__global__ void bond_init_kernel(const float* __restrict__ rbf, const float* __restrict__ W,
                                 const float* __restrict__ b, float* __restrict__ bond) {
  int idx = blockIdx.x * blockDim.x + threadIdx.x;   // E*64 threads, exact
  int e = idx >> 6, u = idx & 63;
  float s = b[u];
  s += rbf[(size_t)e * 3 + 0] * W[0 * U_N + u];
  s += rbf[(size_t)e * 3 + 1] * W[1 * U_N + u];
  s += rbf[(size_t)e * 3 + 2] * W[2 * U_N + u];
  bond[idx] = silu_(s);
}

// ------------------------------------ watom = sigmoid(atom @ tu_W + tu_b) WMMA
// tu_W: 64 x 48. LDS: wt[48][68]
__global__ __launch_bounds__(256)
void tu_kernel(const float* __restrict__ atom, const float* __restrict__ W,
               const float* __restrict__ bv, float* __restrict__ watom, int ntiles) {
  extern __shared__ float smem[];
  const int KP = U_N + 4;                            // 68
  stage_wt(W, smem, U_N, RBF_N, KP);
  __syncthreads();
  const int wave = threadIdx.x >> 5, lane = threadIdx.x & 31;
  const int tile = blockIdx.x * 8 + wave;
  if (tile >= ntiles) return;                        // wave-uniform: EXEC stays all-1
  const int m = lane & 15, hi = lane >> 4;
  const int row = tile * 16;
  const v8f z = {0.f,0.f,0.f,0.f,0.f,0.f,0.f,0.f};
  v8f acc[3] = {z, z, z};
  #pragma unroll
  for (int k = 0; k < U_N; k += 4) {
    const int k0 = k + 2 * hi;
    v2f av = *(const v2f*)(atom + (size_t)(row + m) * U_N + k0);
    #pragma unroll
    for (int n = 0; n < 3; ++n) {
      const int nc = n * 16 + m;
      v2f bf = *(const v2f*)(smem + nc * KP + k0);
      acc[n] = wmma4(av, bf, acc[n]);
    }
  }
  #pragma unroll
  for (int n = 0; n < 3; ++n) {
    const int nc = n * 16 + m;
    const float bb = bv[nc];
    #pragma unroll
    for (int r = 0; r < 8; ++r) {
      const int a = row + r + 8 * hi;
      watom[(size_t)a * RBF_N + nc] = sig_(acc[n][r] + bb);
    }
  }
}

// --------------- agg[tb0] += three_basis * watom[dst[tb1]] (scatter, atomics)
__global__ __launch_bounds__(256)
void triple_kernel(const int* __restrict__ tbi, const int* __restrict__ bai,
                   const float* __restrict__ tb, const float* __restrict__ watom,
                   float* __restrict__ agg) {
  const int tl = threadIdx.x >> 4;                   // 16 triples per block
  const int rg = threadIdx.x & 15;                   // 16 groups of 3 rbf each
  const size_t t = (size_t)blockIdx.x * 16 + tl;
  const int e0 = tbi[2 * t], e1 = tbi[2 * t + 1];
  const int ea = bai[2 * (size_t)e1 + 1];            // end_atom = dst[tb1]
  const float* wrow = watom + (size_t)ea * RBF_N;
  const float* brow = tb + t * RBF_N;
  float* arow = agg + (size_t)e0 * RBF_N;
  #pragma unroll
  for (int j = 0; j < 3; ++j) {
    const int r = rg * 3 + j;
    atomicAdd(arow + r, brow[r] * wrow[r]);
  }
}

// ------------- bond += silu(agg@W1+b1)*sig(agg@W2+b2)   [E x 48 @ 48 x 64] WMMA
// LDS: w1t[64][52], w2t[64][52]
__global__ __launch_bounds__(256)
void tf_kernel(const float* __restrict__ agg,
               const float* __restrict__ W1, const float* __restrict__ b1v,
               const float* __restrict__ W2, const float* __restrict__ b2v,
               float* __restrict__ bond) {
  extern __shared__ float smem[];
  const int KP = RBF_N + 4;                          // 52
  float* w1t = smem;
  float* w2t = smem + U_N * KP;
  stage_wt(W1, w1t, RBF_N, U_N, KP);
  stage_wt(W2, w2t, RBF_N, U_N, KP);
  __syncthreads();
  const int wave = threadIdx.x >> 5, lane = threadIdx.x & 31;
  const int tile = blockIdx.x * 8 + wave;
  const int m = lane & 15, hi = lane >> 4;
  const int row = tile * 16;
  const v8f z = {0.f,0.f,0.f,0.f,0.f,0.f,0.f,0.f};
  v8f a1[4] = {z, z, z, z}, a2[4] = {z, z, z, z};
  const float* aptr = agg + (size_t)(row + m) * RBF_N;
  #pragma unroll
  for (int k = 0; k < RBF_N; k += 4) {
    const int k0 = k + 2 * hi;
    v2f av = *(const v2f*)(aptr + k0);
    #pragma unroll
    for (int n = 0; n < 4; ++n) {
      const int nc = n * 16 + m;
      v2f bf1 = *(const v2f*)(w1t + nc * KP + k0);
      a1[n] = wmma4(av, bf1, a1[n]);
      v2f bf2 = *(const v2f*)(w2t + nc * KP + k0);
      a2[n] = wmma4(av, bf2, a2[n]);
    }
  }
  #pragma unroll
  for (int n = 0; n < 4; ++n) {
    const int nc = n * 16 + m;
    const float bb1 = b1v[nc], bb2 = b2v[nc];
    #pragma unroll
    for (int r = 0; r < 8; ++r) {
      const size_t e = (size_t)(row + r + 8 * hi);
      const float x1 = a1[n][r] + bb1, x2 = a2[n][r] + bb2;
      bond[e * U_N + nc] += silu_(x1) * sig_(x2);
    }
  }
}

// ------- bond += silu(cat@W1+b1)*sig(cat@W2+b2), cat=[atom[src],atom[dst],bond]
// LDS: w1t[64][196], w2t[64][196]
__global__ __launch_bounds__(256)
void gb_kernel(const float* __restrict__ atom, float* __restrict__ bond,
               const int* __restrict__ bai,
               const float* __restrict__ W1, const float* __restrict__ b1v,
               const float* __restrict__ W2, const float* __restrict__ b2v) {
  extern __shared__ float smem[];
  const int K = 3 * U_N, KP = K + 4;                 // 192, 196
  float* w1t = smem;
  float* w2t = smem + U_N * KP;
  stage_wt(W1, w1t, K, U_N, KP);
  stage_wt(W2, w2t, K, U_N, KP);
  __syncthreads();
  const int wave = threadIdx.x >> 5, lane = threadIdx.x & 31;
  const int tile = blockIdx.x * 8 + wave;
  const int m = lane & 15, hi = lane >> 4;
  const int row = tile * 16;
  const int em = row + m;
  const int se = bai[2 * (size_t)em], de = bai[2 * (size_t)em + 1];
  const float* segp[3];
  segp[0] = atom + (size_t)se * U_N;                 // cat[:,  0.. 63] = atom[src]
  segp[1] = atom + (size_t)de * U_N;                 // cat[:, 64..127] = atom[dst]
  segp[2] = bond + (size_t)em * U_N;                 // cat[:,128..191] = bond
  const v8f z = {0.f,0.f,0.f,0.f,0.f,0.f,0.f,0.f};
  v8f a1[4] = {z, z, z, z}, a2[4] = {z, z, z, z};
  for (int k = 0; k < K; k += 4) {
    const int k0 = k + 2 * hi;                       // even; never crosses a 64-seg
    v2f av = *(const v2f*)(segp[k0 >> 6] + (k0 & 63));
    #pragma unroll
    for (int n = 0; n < 4; ++n) {
      const int nc = n * 16 + m;
      v2f bf1 = *(const v2f*)(w1t + nc * KP + k0);
      a1[n] = wmma4(av, bf1, a1[n]);
      v2f bf2 = *(const v2f*)(w2t + nc * KP + k0);
      a2[n] = wmma4(av, bf2, a2[n]);
    }
  }
  #pragma unroll
  for (int n = 0; n < 4; ++n) {
    const int nc = n * 16 + m;
    const float bb1 = b1v[nc], bb2 = b2v[nc];
    #pragma unroll
    for (int r = 0; r < 8; ++r) {
      const size_t e = (size_t)(row + r + 8 * hi);
      const float x1 = a1[n][r] + bb1, x2 = a2[n][r] + bb2;
      bond[e * U_N + nc] += silu_(x1) * sig_(x2);
    }
  }
}

// -------- msg = silu(bond@W1+b1)*sig(bond@W2+b2); atom[dst] += msg (atomics)
// LDS: w1t[64][68], w2t[64][68]
__global__ __launch_bounds__(256)
void ga_kernel(const float* __restrict__ bond, const int* __restrict__ bai,
               const float* __restrict__ W1, const float* __restrict__ b1v,
               const float* __restrict__ W2, const float* __restrict__ b2v,
               float* __restrict__ atom) {
  extern __shared__ float smem[];
  const int KP = U_N + 4;                            // 68
  float* w1t = smem;
  float* w2t = smem + U_N * KP;
  stage_wt(W1, w1t, U_N, U_N, KP);
  stage_wt(W2, w2t, U_N, U_N, KP);
  __syncthreads();
  const int wave = threadIdx.x >> 5, lane = threadIdx.x & 31;
  const int tile = blockIdx.x * 8 + wave;
  const int m = lane & 15, hi = lane >> 4;
  const int row = tile * 16;
  const v8f z = {0.f,0.f,0.f,0.f,0.f,0.f,0.f,0.f};
  v8f a1[4] = {z, z, z, z}, a2[4] = {z, z, z, z};
  const float* aptr = bond + (size_t)(row + m) * U_N;
  #pragma unroll
  for (int k = 0; k < U_N; k += 4) {
    const int k0 = k + 2 * hi;
    v2f av = *(const v2f*)(aptr + k0);
    #pragma unroll
    for (int n = 0; n < 4; ++n) {
      const int nc = n * 16 + m;
      v2f bf1 = *(const v2f*)(w1t + nc * KP + k0);
      a1[n] = wmma4(av, bf1, a1[n]);
      v2f bf2 = *(const v2f*)(w2t + nc * KP + k0);
      a2[n] = wmma4(av, bf2, a2[n]);
    }
  }
  #pragma unroll
  for (int r = 0; r < 8; ++r) {
    const int e = row + r + 8 * hi;
    const int d = bai[2 * (size_t)e + 1];
    #pragma unroll
    for (int n = 0; n < 4; ++n) {
      const int nc = n * 16 + m;
      const float x1 = a1[n][r] + b1v[nc], x2 = a2[n][r] + b2v[nc];
      atomicAdd(atom + (size_t)d * U_N + nc, silu_(x1) * sig_(x2));
    }
  }
}

// ------------------------- gated mean pooling:  num[g] += sig(atom@wrW+b)*atom
__global__ __launch_bounds__(64)
void readout_pool(const float* __restrict__ atom, const int* __restrict__ batch,
                  const float* __restrict__ wrW, const float* __restrict__ wrb,
                  float* __restrict__ num, float* __restrict__ cnt) {
  __shared__ float row[U_N];
  const int a = blockIdx.x, u = threadIdx.x;
  row[u] = atom[(size_t)a * U_N + u];
  __syncthreads();
  float g = wrb[u];
  #pragma unroll 8
  for (int k = 0; k < U_N; ++k) g += row[k] * wrW[k * U_N + u];
  const float gate = sig_(g);
  const int b = batch[a];
  atomicAdd(num + (size_t)b * U_N + u, gate * row[u]);
  if (u == 0) atomicAdd(cnt + b, 1.0f);
}

// -------------------------------- out = silu(vec@W1+b1)@W2 + b2  per graph
__global__ __launch_bounds__(64)
void final_kernel(const float* __restrict__ num, const float* __restrict__ cnt,
                  const float* __restrict__ W1, const float* __restrict__ b1,
                  const float* __restrict__ W2, const float* __restrict__ b2,
                  float* __restrict__ out) {
  __shared__ float vec[U_N];
  __shared__ float red[U_N];
  const int g = blockIdx.x, u = threadIdx.x;
  const float c = fmaxf(cnt[g], 1.0f);
  vec[u] = num[(size_t)g * U_N + u] / c;
  __syncthreads();
  float h = b1[u];
  #pragma unroll 8
  for (int k = 0; k < U_N; ++k) h += vec[k] * W1[k * U_N + u];
  red[u] = silu_(h) * W2[u];
  __syncthreads();
  for (int s = 32; s > 0; s >>= 1) {
    if (u < s) red[u] += red[u + s];
    __syncthreads();
  }
  if (u == 0) out[g] = red[0] + b2[0];
}

extern "C" void kernel_launch(void* const* d_in, const int* in_sizes, int n_in,
                              void* d_out, int out_size, void* d_ws, size_t ws_size,
                              hipStream_t stream) {
  (void)in_sizes; (void)n_in; (void)out_size; (void)ws_size;
  const int*   atom_types = (const int*)d_in[0];
  const int*   bai        = (const int*)d_in[1];
  const int*   tbi        = (const int*)d_in[2];
  const int*   batch      = (const int*)d_in[3];
  const float* bond_rbf   = (const float*)d_in[4];
  const float* three_b    = (const float*)d_in[5];
  const float* emb        = (const float*)d_in[6];
  const float* bond_W     = (const float*)d_in[7];
  const float* bond_b     = (const float*)d_in[8];
  const float* tu_W       = (const float*)d_in[9];
  const float* tu_b       = (const float*)d_in[10];
  const float* tf_W1      = (const float*)d_in[11];
  const float* tf_b1      = (const float*)d_in[12];
  const float* tf_W2      = (const float*)d_in[13];
  const float* tf_b2      = (const float*)d_in[14];
  const float* gb_W1      = (const float*)d_in[15];
  const float* gb_b1      = (const float*)d_in[16];
  const float* gb_W2      = (const float*)d_in[17];
  const float* gb_b2      = (const float*)d_in[18];
  const float* ga_W1      = (const float*)d_in[19];
  const float* ga_b1      = (const float*)d_in[20];
  const float* ga_W2      = (const float*)d_in[21];
  const float* ga_b2      = (const float*)d_in[22];
  const float* wr_W       = (const float*)d_in[23];
  const float* wr_b       = (const float*)d_in[24];
  const float* f_W1       = (const float*)d_in[25];
  const float* f_b1       = (const float*)d_in[26];
  const float* f_W2       = (const float*)d_in[27];
  const float* f_b2       = (const float*)d_in[28];

  // workspace layout (fp32): atom | watom | agg | bond | num | cnt
  float* atom  = (float*)d_ws;
  float* watom = atom  + (size_t)A_N * U_N;
  float* agg   = watom + (size_t)A_N * RBF_N;
  float* bond  = agg   + (size_t)E_N * RBF_N;
  float* num   = bond  + (size_t)E_N * U_N;
  float* cnt   = num   + (size_t)G_N * U_N;

  // dynamic LDS sizes (transposed+padded weight tiles)
  const size_t lds_tu = sizeof(float) * (size_t)RBF_N * (U_N + 4);       // 48*68
  const size_t lds_tf = sizeof(float) * 2u * U_N * (RBF_N + 4);          // 2*64*52
  const size_t lds_gb = sizeof(float) * 2u * U_N * (3 * U_N + 4);        // 2*64*196
  const size_t lds_ga = sizeof(float) * 2u * U_N * (U_N + 4);            // 2*64*68

  embed_kernel<<<(A_N * U_N) / 256, 256, 0, stream>>>(atom_types, emb, atom);
  bond_init_kernel<<<(E_N * U_N) / 256, 256, 0, stream>>>(bond_rbf, bond_W, bond_b, bond);

  const int atiles = A_N / 16;                       // 3125
  const int etiles = E_N / 16;                       // 75000 (divisible by 8)
  for (int i = 0; i < 3; ++i) {
    tu_kernel<<<(atiles + 7) / 8, 256, lds_tu, stream>>>(
        atom, tu_W + (size_t)i * U_N * RBF_N, tu_b + (size_t)i * RBF_N, watom, atiles);
    hipMemsetAsync(agg, 0, (size_t)E_N * RBF_N * sizeof(float), stream);
    triple_kernel<<<T_N / 16, 256, 0, stream>>>(tbi, bai, three_b, watom, agg);
    tf_kernel<<<etiles / 8, 256, lds_tf, stream>>>(
        agg, tf_W1 + (size_t)i * RBF_N * U_N, tf_b1 + (size_t)i * U_N,
             tf_W2 + (size_t)i * RBF_N * U_N, tf_b2 + (size_t)i * U_N, bond);
    gb_kernel<<<etiles / 8, 256, lds_gb, stream>>>(
        atom, bond, bai,
        gb_W1 + (size_t)i * 3 * U_N * U_N, gb_b1 + (size_t)i * U_N,
        gb_W2 + (size_t)i * 3 * U_N * U_N, gb_b2 + (size_t)i * U_N);
    ga_kernel<<<etiles / 8, 256, lds_ga, stream>>>(
        bond, bai,
        ga_W1 + (size_t)i * U_N * U_N, ga_b1 + (size_t)i * U_N,
        ga_W2 + (size_t)i * U_N * U_N, ga_b2 + (size_t)i * U_N, atom);
  }

  hipMemsetAsync(num, 0, (size_t)(G_N * U_N + G_N) * sizeof(float), stream);
  readout_pool<<<A_N, 64, 0, stream>>>(atom, batch, wr_W, wr_b, num, cnt);
  final_kernel<<<G_N, 64, 0, stream>>>(num, cnt, f_W1, f_b1, f_W2, f_b2, (float*)d_out);
}